// ProxyLabelLoss_26749056320202
// MI455X (gfx1250) — compile-verified
//
#include <hip/hip_runtime.h>
#include <stdint.h>

// Problem shapes (fixed by setup_inputs in the reference)
#define B_  8
#define H_  512
#define W_  1024
#define hF  128
#define wF  256

#define CNT_DX (8.0f * 2.0f * 128.0f * 255.0f)   // 522240
#define CNT_DY (8.0f * 2.0f * 127.0f * 256.0f)   // 520192

#if defined(__has_builtin)
# if __has_builtin(__builtin_amdgcn_tensor_load_to_lds) && __has_builtin(__builtin_amdgcn_s_wait_tensorcnt)
#  define USE_TDM 1
# endif
# if __has_builtin(__builtin_amdgcn_wmma_f32_16x16x4_f32)
#  define USE_WMMA_REDUCE 1
# endif
#endif

typedef unsigned int v4u __attribute__((ext_vector_type(4)));
typedef int          v8i __attribute__((ext_vector_type(8)));
typedef int          v4i __attribute__((ext_vector_type(4)));
typedef float        v2f __attribute__((ext_vector_type(2)));
typedef float        v8f __attribute__((ext_vector_type(8)));

__device__ __forceinline__ float bilin(const float* __restrict__ p,
                                       int r0, int r1, int c0, int c1,
                                       float wr, float wc, int pitch) {
  float v00 = p[(size_t)r0 * pitch + c0];
  float v01 = p[(size_t)r0 * pitch + c1];
  float v10 = p[(size_t)r1 * pitch + c0];
  float v11 = p[(size_t)r1 * pitch + c1];
  float top = v00 + (v01 - v00) * wc;
  float bot = v10 + (v11 - v10) * wc;
  return top + (bot - top) * wr;
}

__global__ void zero_acc_kernel(float* __restrict__ acc) {
  if (threadIdx.x < 4) acc[threadIdx.x] = 0.0f;
}

// ---------------- smoothness: sum |dx| into acc[2], sum |dy| into acc[3] ----
__global__ __launch_bounds__(256) void smooth_kernel(const float* __restrict__ flow,
                                                     float* __restrict__ acc) {
  const int i = blockIdx.x * 256 + threadIdx.x;     // over B*2*h*w = 524288
  float sdx = 0.0f, sdy = 0.0f;
  if (i < B_ * 2 * hF * wF) {
    const int x = i & (wF - 1);
    const int y = (i >> 8) & (hF - 1);
    const float v = flow[i];
    if (x < wF - 1) sdx = fabsf(v - flow[i + 1]);
    if (y < hF - 1) sdy = fabsf(v - flow[i + wF]);
  }
  #pragma unroll
  for (int o = 16; o > 0; o >>= 1) {
    sdx += __shfl_down(sdx, o, 32);
    sdy += __shfl_down(sdy, o, 32);
  }
  __shared__ float rx[8], ry[8];
  const int lane = threadIdx.x & 31, wv = threadIdx.x >> 5;
  if (lane == 0) { rx[wv] = sdx; ry[wv] = sdy; }
  __syncthreads();
  if (threadIdx.x == 0) {
    float a = 0.0f, b = 0.0f;
    #pragma unroll
    for (int k = 0; k < 8; ++k) { a += rx[k]; b += ry[k]; }
    atomicAdd(&acc[2], a);
    atomicAdd(&acc[3], b);
  }
}

// ---------------- fused warp + SSIM + L1 + masked reduction -----------------
// Block: 16x16 threads -> one 16x16 output tile; 18x18 haloed LDS tiles.
// img1 tile arrives via TDM (tensor_load_to_lds) while threads compute the
// warped img2 tile into LDS; then the 3x3 stencils run entirely from LDS.
// Block reduction runs on the matrix pipe: V_WMMA_F32_16X16X4_F32 with
// B = ones computes row-sums; 4 chained accumulating WMMAs fold 256 partials.
__global__ __launch_bounds__(256) void FlowPhotoLoss_fused_kernel(
    const float* __restrict__ flow, const float* __restrict__ img1,
    const float* __restrict__ img2, const float* __restrict__ mask,
    float* __restrict__ acc)
{
  __shared__ float s_i1[3 * 18 * 18];   // compacted [c][row][col], pitch tw
  __shared__ float s_w [3 * 18 * 19];   // warped img2, fixed pitch 19
  __shared__ float s_pn[256], s_pd[256];

  const int bx = blockIdx.x, by = blockIdx.y, b = blockIdx.z;
  const int tid  = threadIdx.y * 16 + threadIdx.x;
  const int x0b  = bx * 16 - 1,  y0b = by * 16 - 1;     // tile origin incl. halo
  const int xoff = (x0b < 0) ? 1 : 0;
  const int yoff = (y0b < 0) ? 1 : 0;
  const int gx0  = x0b + xoff,  gy0 = y0b + yoff;       // clamped in-image origin
  const int tw   = 18 - xoff,   th  = 18 - yoff;        // transferred tile dims

  const int gx = bx * 16 + threadIdx.x;                 // center pixel (always in-image)
  const int gy = by * 16 + threadIdx.y;
  __builtin_prefetch(&mask[((size_t)b * H_ + gy) * W_ + gx], 0, 0);

#ifdef USE_TDM
  // One 3D TDM descriptor: tile = tw x th x 3 channels of img1, OOB right/bottom
  // reads return 0 (matches the zero-padded avg_pool). LDS dest is contiguous:
  // offset = ((c*th + r)*tw + col)*4.
  if (tid == 0) {
    const uint64_t ga = (uint64_t)(uintptr_t)(img1 + (((size_t)b * 3 + 0) * H_ + gy0) * W_ + gx0);
    const uint32_t lds = (uint32_t)(uintptr_t)&s_i1[0];
    const uint32_t td0 = (uint32_t)(W_ - gx0);          // dim0 extent from tile start
    const uint32_t td1 = (uint32_t)(H_ - gy0);          // dim1 extent from tile start
    v4u g0 = { 1u,                                      // count=1 valid descriptor
               lds,                                     // lds_addr
               (uint32_t)ga,                            // global_addr[31:0]
               (uint32_t)((ga >> 32) & 0x1FFFFFFu) | (2u << 30) }; // addr[56:32] | type=2
    v8i g1 = { (int)0x00020000u,                        // wg_mask=0, data_size=2 (4B)
               (int)((td0 & 0xFFFFu) << 16),            // tensor_dim0 lo
               (int)((td0 >> 16) | ((td1 & 0xFFFFu) << 16)),
               (int)((td1 >> 16) | ((uint32_t)tw << 16)),   // tile_dim0 = tw
               (int)((uint32_t)th | (3u << 16)),        // tile_dim1 = th, tile_dim2 = 3
               (int)(uint32_t)W_,                       // tensor_dim0_stride lo (1024)
               (int)0,                                  // stride0 hi | stride1[15:0]
               (int)((uint32_t)(H_ * W_) >> 16) };      // tensor_dim1_stride (524288) hi
    v4i g2 = { 3, 0, 0, 0 };                            // tensor_dim2 = 3 channels
    v4i g3 = { 0, 0, 0, 0 };
    v8i g4 = { 0, 0, 0, 0, 0, 0, 0, 0 };                // extra group (clang-23 form)
    __builtin_amdgcn_tensor_load_to_lds(g0, g1, g2, g3, g4, 0);
  }
#else
  // Fallback: cooperative loads into the same compacted layout.
  for (int idx = tid; idx < 3 * th * tw; idx += 256) {
    const int c   = idx / (th * tw);
    const int rem = idx - c * th * tw;
    const int r   = rem / tw;
    const int col = rem - r * tw;
    const int yy = gy0 + r, xx = gx0 + col;
    float v = 0.0f;
    if (yy < H_ && xx < W_) v = img1[(((size_t)b * 3 + c) * H_ + yy) * W_ + xx];
    s_i1[(c * th + r) * tw + col] = v;
  }
#endif

  // ---- overlapped with the TDM DMA: compute warped img2 tile into LDS ----
  const float RY = (float)(hF - 1) / (float)(H_ - 1);
  const float RX = (float)(wF - 1) / (float)(W_ - 1);
  const float* f0 = flow + ((size_t)b * 2 + 0) * hF * wF;
  const float* f1 = flow + ((size_t)b * 2 + 1) * hF * wF;
  for (int idx = tid; idx < 18 * 18; idx += 256) {
    const int ty = idx / 18, tx = idx - ty * 18;
    const int yy = y0b + ty, xx = x0b + tx;
    float wv0 = 0.0f, wv1 = 0.0f, wv2 = 0.0f;        // zero pad outside image
    if (yy >= 0 && yy < H_ && xx >= 0 && xx < W_) {
      // flow upsample (bilinear, align_corners) then scale by 4 in each dim
      const float yc = yy * RY, xc = xx * RX;
      const int fy0 = (int)yc; const int fy1 = min(fy0 + 1, hF - 1); const float wyf = yc - fy0;
      const int fx0 = (int)xc; const int fx1 = min(fx0 + 1, wF - 1); const float wxf = xc - fx0;
      const float flx = bilin(f0, fy0, fy1, fx0, fx1, wyf, wxf, wF) * (float)(W_ / wF);
      const float fly = bilin(f1, fy0, fy1, fx0, fx1, wyf, wxf, wF) * (float)(H_ / hF);
      // grid_sample border+align_corners collapses to clip(p + flow, 0, dim-1)
      const float sx = fminf(fmaxf((float)xx + flx, 0.0f), (float)(W_ - 1));
      const float sy = fminf(fmaxf((float)yy + fly, 0.0f), (float)(H_ - 1));
      const int ix0 = (int)sx; const int ix1 = min(ix0 + 1, W_ - 1); const float wxs = sx - ix0;
      const int iy0 = (int)sy; const int iy1 = min(iy0 + 1, H_ - 1); const float wys = sy - iy0;
      const float* p0 = img2 + (((size_t)b * 3 + 0) * H_) * W_;
      const float* p1 = img2 + (((size_t)b * 3 + 1) * H_) * W_;
      const float* p2 = img2 + (((size_t)b * 3 + 2) * H_) * W_;
      wv0 = bilin(p0, iy0, iy1, ix0, ix1, wys, wxs, W_);
      wv1 = bilin(p1, iy0, iy1, ix0, ix1, wys, wxs, W_);
      wv2 = bilin(p2, iy0, iy1, ix0, ix1, wys, wxs, W_);
    }
    s_w[(0 * 18 + ty) * 19 + tx] = wv0;
    s_w[(1 * 18 + ty) * 19 + tx] = wv1;
    s_w[(2 * 18 + ty) * 19 + tx] = wv2;
  }

#ifdef USE_TDM
  __builtin_amdgcn_s_wait_tensorcnt(0);   // TDM tile resident before barrier
#endif
  __syncthreads();

  // ---- 3x3 zero-padded box stats from LDS; SSIM + L1 per channel ----
  const int tlx = threadIdx.x + 1, tly = threadIdx.y + 1;   // tile-local center
  const float C1 = 0.0001f, C2 = 0.0009f, inv9 = 1.0f / 9.0f;
  float ssim_acc = 0.0f, l1_acc = 0.0f;
  #pragma unroll
  for (int c = 0; c < 3; ++c) {
    float sa = 0.f, sb = 0.f, saa = 0.f, sbb = 0.f, sab = 0.f;
    #pragma unroll
    for (int dy2 = -1; dy2 <= 1; ++dy2) {
      #pragma unroll
      for (int dx2 = -1; dx2 <= 1; ++dx2) {
        const int ty = tly + dy2, tx = tlx + dx2;
        float a = 0.0f;
        if (ty >= yoff && tx >= xoff)
          a = s_i1[(c * th + (ty - yoff)) * tw + (tx - xoff)];
        const float w = s_w[(c * 18 + ty) * 19 + tx];
        sa += a; sb += w; saa += a * a; sbb += w * w; sab += a * w;
      }
    }
    const float mux = sa * inv9, muy = sb * inv9;
    const float sxx = saa * inv9 - mux * mux;
    const float syy = sbb * inv9 - muy * muy;
    const float sxy = sab * inv9 - mux * muy;
    const float ssim = (2.0f * mux * muy + C1) * (2.0f * sxy + C2) /
                       ((mux * mux + muy * muy + C1) * (sxx + syy + C2) + 1e-12f);
    ssim_acc += fminf(fmaxf((1.0f - ssim) * 0.5f, 0.0f), 1.0f);
    const float ac = s_i1[(c * th + (tly - yoff)) * tw + (tlx - xoff)];
    const float bc = s_w[(c * 18 + tly) * 19 + tlx];
    l1_acc += fabsf(ac - bc);
  }
  const float photo_map = 0.85f * (ssim_acc * (1.0f / 3.0f)) +
                          0.15f * (l1_acc * (1.0f / 3.0f));
  const float m = (mask[((size_t)b * H_ + gy) * W_ + gx] > 0.5f) ? 1.0f : 0.0f;
  const float num = photo_map * m;
  const float den = m;

#ifdef USE_WMMA_REDUCE
  // ---- block reduction on the matrix pipe ----
  // D = A x ones(4x16) + C gives row sums of A; summing one column of D over
  // all 16 rows yields the total of all 64 A elements (any packing is valid).
  s_pn[tid] = num;
  s_pd[tid] = den;
  __syncthreads();
  if (tid < 32) {                       // exactly wave 0, EXEC all ones
    const v2f ones = { 1.0f, 1.0f };    // B = 4x16 all-ones
    v8f cn = { 0.f, 0.f, 0.f, 0.f, 0.f, 0.f, 0.f, 0.f };
    v8f cd = { 0.f, 0.f, 0.f, 0.f, 0.f, 0.f, 0.f, 0.f };
    #pragma unroll
    for (int ch = 0; ch < 4; ++ch) {    // 4 chunks of 64 partials each
      v2f an = { s_pn[ch * 64 + tid * 2], s_pn[ch * 64 + tid * 2 + 1] };
      v2f ad = { s_pd[ch * 64 + tid * 2], s_pd[ch * 64 + tid * 2 + 1] };
      cn = __builtin_amdgcn_wmma_f32_16x16x4_f32(false, an, false, ones,
                                                 (short)0, cn, false, false);
      cd = __builtin_amdgcn_wmma_f32_16x16x4_f32(false, ad, false, ones,
                                                 (short)0, cd, false, false);
    }
    // lanes 0..15 hold rows 0..7 of column (lane); lanes 16..31 hold rows 8..15
    float sn = cn[0]+cn[1]+cn[2]+cn[3]+cn[4]+cn[5]+cn[6]+cn[7];
    float sd = cd[0]+cd[1]+cd[2]+cd[3]+cd[4]+cd[5]+cd[6]+cd[7];
    sn += __shfl(sn, 16, 32);           // add other half's rows (column 0)
    sd += __shfl(sd, 16, 32);
    if (tid == 0) {
      atomicAdd(&acc[0], sn);
      atomicAdd(&acc[1], sd);
    }
  }
#else
  // ---- fallback: wave32 shfl + LDS reduction ----
  float n2 = num, d2 = den;
  #pragma unroll
  for (int o = 16; o > 0; o >>= 1) {
    n2 += __shfl_down(n2, o, 32);
    d2 += __shfl_down(d2, o, 32);
  }
  const int lane = tid & 31, wv = tid >> 5;
  if (lane == 0) { s_pn[wv] = n2; s_pd[wv] = d2; }
  __syncthreads();
  if (tid == 0) {
    float n = 0.0f, d = 0.0f;
    #pragma unroll
    for (int k = 0; k < 8; ++k) { n += s_pn[k]; d += s_pd[k]; }
    atomicAdd(&acc[0], n);
    atomicAdd(&acc[1], d);
  }
#endif
}

// ---------------- finalize: (total, photo, smooth) --------------------------
__global__ void finalize_kernel(const float* __restrict__ acc, float* __restrict__ out) {
  const float photo  = acc[0] / fmaxf(acc[1], 1.0f);
  const float smooth = acc[2] * (1.0f / CNT_DX) + acc[3] * (1.0f / CNT_DY);
  const float total  = photo + 0.1f * smooth;
  out[0] = total;
  out[1] = photo;
  out[2] = smooth;
}

extern "C" void kernel_launch(void* const* d_in, const int* in_sizes, int n_in,
                              void* d_out, int out_size, void* d_ws, size_t ws_size,
                              hipStream_t stream) {
  (void)in_sizes; (void)n_in; (void)out_size; (void)ws_size;
  const float* flow = (const float*)d_in[0];
  const float* img1 = (const float*)d_in[1];
  const float* img2 = (const float*)d_in[2];
  const float* mask = (const float*)d_in[3];
  float* out = (float*)d_out;
  float* acc = (float*)d_ws;   // [photo_num, photo_den, sum|dx|, sum|dy|]

  zero_acc_kernel<<<1, 32, 0, stream>>>(acc);
  smooth_kernel<<<(B_ * 2 * hF * wF + 255) / 256, 256, 0, stream>>>(flow, acc);
  FlowPhotoLoss_fused_kernel<<<dim3(W_ / 16, H_ / 16, B_), dim3(16, 16), 0, stream>>>(
      flow, img1, img2, mask, acc);
  finalize_kernel<<<1, 1, 0, stream>>>(acc, out);
}